// MultiAgentsSummarizer_42726334660798
// MI455X (gfx1250) — compile-verified
//
#include <hip/hip_runtime.h>

// dims: T=tgt_len, B=batch, A=agents, S=src_len, V=vocab, N_OOV extra slots
constexpr int T_DIM = 64, B_DIM = 8, A_DIM = 3, S_DIM = 400;
constexpr int V_DIM = 50000, N_OOV = 400, V_EXT = V_DIM + N_OOV;
constexpr int AS    = A_DIM * S_DIM;   // 1200 scatter entries per (t,b)
constexpr int BLOCK = 256;             // 8 wave32 waves
constexpr int NV4   = V_DIM / 4;       // 12500 float4 per row (base vocab)
constexpr int NE4   = V_EXT / 4;       // 12600 float4 per row (extended)

#if defined(__HIP_DEVICE_COMPILE__)
#if __has_builtin(__builtin_amdgcn_global_load_async_to_lds_b32)
#define HAVE_ASYNC_LDS 1
#endif
#endif

typedef __attribute__((address_space(1))) int g_i32;  // global (AS1)
typedef __attribute__((address_space(3))) int l_i32;  // LDS (AS3)

__device__ __forceinline__ void async_g2l_b32(const void* g, void* l) {
#ifdef HAVE_ASYNC_LDS
  // gfx1250 async DMA: global -> LDS, tracked by ASYNCcnt.
  // Builtin expects (AS1 int*, AS3 int*, imm offset, imm cpol).
  __builtin_amdgcn_global_load_async_to_lds_b32(
      (g_i32*)g, (l_i32*)l, /*imm offset=*/0, /*cpol=*/0);
#else
  *(int*)l = *(const int*)g;  // synchronous fallback
#endif
}

__device__ __forceinline__ void wait_async_zero() {
#ifdef HAVE_ASYNC_LDS
#if __has_builtin(__builtin_amdgcn_s_wait_asynccnt)
  __builtin_amdgcn_s_wait_asynccnt(0);
#else
  asm volatile("s_wait_asynccnt 0" ::: "memory");
#endif
#endif
}

// One workgroup owns one (t,b) output row of V_EXT floats.
__global__ __launch_bounds__(BLOCK) void MultiAgentsSummarizer_42726334660798_kernel(
    const int*   __restrict__ article,      // [S,B,A]
    const float* __restrict__ vocab_probs,  // [T,B,V]
    const float* __restrict__ gen_probs,    // [T,B,A]
    const float* __restrict__ agw_attn,     // [T,B,A,S]
    const float* __restrict__ agent_attn,   // [T,B,A]
    float*       __restrict__ out)          // [T,B,V_EXT]
{
  __shared__ float s_attn[AS];  // agentwise_attn row, layout k = a*S + s
  __shared__ int   s_tok[AS];   // gathered article tokens, same layout

  const int tb  = blockIdx.x;        // t*B + b
  const int b   = tb % B_DIM;
  const int tid = threadIdx.x;

  // ---- Phase 0: kick off async staging of scatter operands into LDS.
  // These DMAs run while Phase 1 saturates HBM with the streaming row init.
  const float* attn_src = agw_attn + (size_t)tb * AS;  // contiguous 1200 floats
  for (int k = tid; k < AS; k += BLOCK)
    async_g2l_b32(attn_src + k, s_attn + k);
  for (int k = tid; k < AS; k += BLOCK) {
    const int a = k / S_DIM;
    const int s = k - a * S_DIM;
    // article[s,b,a] -> s_tok[a*S+s]
    async_g2l_b32(article + ((size_t)s * B_DIM + b) * A_DIM + a, s_tok + k);
  }

  // ---- Per-(t,b) scalars (6 floats; computed redundantly by every thread)
  float coeff = 0.0f, w0, w1, w2;
  {
    float w[A_DIM];
#pragma unroll
    for (int a = 0; a < A_DIM; ++a) {
      const float g  = gen_probs[tb * A_DIM + a];
      const float at = agent_attn[tb * A_DIM + a];
      coeff += at * g;            // Σ_a agent_attn * gen  (generation mix)
      w[a] = at * (1.0f - g);     // per-agent copy weight
    }
    w0 = w[0]; w1 = w[1]; w2 = w[2];
  }

  // ---- Phase 1: stream the output row (bandwidth-bound part).
  // out[tb, v<V] = coeff * vocab_probs[tb, v];  out[tb, v>=V] = 0
  const float4* __restrict__ vp = (const float4*)(vocab_probs + (size_t)tb * V_DIM);
  float*        __restrict__ orow = out + (size_t)tb * V_EXT;
  float4*       __restrict__ op = (float4*)orow;
  for (int i = tid; i < NV4; i += BLOCK) {
    float4 v = vp[i];
    v.x *= coeff; v.y *= coeff; v.z *= coeff; v.w *= coeff;
    op[i] = v;
  }
  for (int i = NV4 + tid; i < NE4; i += BLOCK)
    op[i] = make_float4(0.0f, 0.0f, 0.0f, 0.0f);

  __threadfence();    // row stores visible at the atomic point-of-coherency (L2)
  wait_async_zero();  // ASYNCcnt == 0: LDS staging complete
  __syncthreads();

  // ---- Phase 2: scatter-add the copy distribution (1200 f32 atomics per row).
  for (int k = tid; k < AS; k += BLOCK) {
    const int a = k / S_DIM;
    const float wa = (a == 0) ? w0 : ((a == 1) ? w1 : w2);
    atomicAdd(orow + s_tok[k], wa * s_attn[k]);
  }
}

extern "C" void kernel_launch(void* const* d_in, const int* in_sizes, int n_in,
                              void* d_out, int out_size, void* d_ws, size_t ws_size,
                              hipStream_t stream) {
  (void)in_sizes; (void)n_in; (void)out_size; (void)d_ws; (void)ws_size;
  const int*   article     = (const int*)  d_in[0];   // [S,B,A]
  const float* vocab_probs = (const float*)d_in[1];   // [T,B,V]
  const float* gen_probs   = (const float*)d_in[2];   // [T,B,A]
  const float* agw_attn    = (const float*)d_in[3];   // [T,B,A,S]
  const float* agent_attn  = (const float*)d_in[4];   // [T,B,A]
  float*       out         = (float*)d_out;           // [T,B,V_EXT]

  dim3 grid(T_DIM * B_DIM);   // 512 workgroups, one per (t,b) row
  dim3 block(BLOCK);
  MultiAgentsSummarizer_42726334660798_kernel<<<grid, block, 0, stream>>>(
      article, vocab_probs, gen_probs, agw_attn, agent_attn, out);
}